// GCN_69363721830621
// MI455X (gfx1250) — compile-verified
//
#include <hip/hip_runtime.h>
#include <hip/hip_bf16.h>

typedef __attribute__((ext_vector_type(2))) float v2f;
typedef __attribute__((ext_vector_type(8))) float v8f;

#define FEAT 128   // IN_DIM == HID_DIM == 128
#define OUTD 10

// ---------------------------------------------------------------------------
// H = X (M x 128) @ W (128 x 128), fp32 via V_WMMA_F32_16X16X4_F32.
// One wave per 16x16 output tile; 8 waves/block cover all 8 N-tiles.
// grid.x = M/16. No bias (reference adds bias after aggregation).
// ---------------------------------------------------------------------------
__global__ __launch_bounds__(256) void gcn_gemm128_wmma(
    const float* __restrict__ X, const float* __restrict__ W,
    float* __restrict__ H) {
  const int lane = threadIdx.x & 31;
  const int wave = threadIdx.x >> 5;
  const int row0 = blockIdx.x << 4;   // M tile
  const int col0 = wave << 4;         // N tile (8 waves -> 128 cols)
  const int l16  = lane & 15;
  const int hi   = lane >> 4;         // 0 for lanes 0-15, 1 for lanes 16-31
  const int koff = hi << 1;           // K offset within 4-step: 0 or 2

  const float* __restrict__ xrow = X + (size_t)(row0 + l16) * FEAT;

  v8f c = {};
#pragma unroll 8
  for (int k0 = 0; k0 < FEAT; k0 += 4) {
    // A 16x4: lane holds M=lane%16, K = k0+koff, k0+koff+1 (contiguous -> v2f)
    v2f a = *(const v2f*)(xrow + k0 + koff);
    // B 4x16: lane holds N=lane%16, rows K=k0+koff, k0+koff+1
    v2f b;
    b.x = W[(size_t)(k0 + koff) * FEAT + col0 + l16];
    b.y = W[(size_t)(k0 + koff + 1) * FEAT + col0 + l16];
    c = __builtin_amdgcn_wmma_f32_16x16x4_f32(false, a, false, b,
                                              (short)0, c, false, false);
  }
  // C/D layout: VGPR v, lanes 0-15 -> M=v ; lanes 16-31 -> M=v+8 ; N=lane%16
  const int mbase = row0 + (hi << 3);
#pragma unroll
  for (int v = 0; v < 8; ++v)
    H[(size_t)(mbase + v) * FEAT + col0 + l16] = c[v];
}

// ---------------------------------------------------------------------------
// Degree / normalization
// ---------------------------------------------------------------------------
__global__ void gcn_deg_init(float* __restrict__ deg, int n) {
  int i = blockIdx.x * blockDim.x + threadIdx.x;
  if (i < n) deg[i] = 1.0f;           // self-loop contribution
}

__global__ void gcn_deg_count(const long long* __restrict__ ei,
                              float* __restrict__ deg, int E) {
  int e = blockIdx.x * blockDim.x + threadIdx.x;
  if (e < E) atomicAdd(&deg[(int)ei[(size_t)E + e]], 1.0f);  // col = ei[1][e]
}

__global__ void gcn_dinv(const float* __restrict__ deg,
                         float* __restrict__ dinv, int n) {
  int i = blockIdx.x * blockDim.x + threadIdx.x;
  if (i < n) dinv[i] = rsqrtf(deg[i]);  // deg >= 1 always (self loops)
}

// ---------------------------------------------------------------------------
// Zero an fp32 buffer
// ---------------------------------------------------------------------------
__global__ void gcn_zero(float* __restrict__ p, int total) {
  int i = blockIdx.x * blockDim.x + threadIdx.x;
  if (i < total) p[i] = 0.0f;
}

// ---------------------------------------------------------------------------
// Edge aggregation: acc[col] += H[row] * (dinv[row]*dinv[col]).
// One wave per edge (E real edges + N self loops); each lane owns 4 floats.
// ---------------------------------------------------------------------------
__global__ __launch_bounds__(256) void gcn_aggregate(
    const float* __restrict__ H, const float* __restrict__ dinv,
    const long long* __restrict__ ei, float* __restrict__ acc,
    int E, int n) {
  const int wgl  = blockIdx.x * (blockDim.x >> 5) + (threadIdx.x >> 5);
  const int lane = threadIdx.x & 31;
  if (wgl >= E + n) return;
  int r, c;
  if (wgl < E) {
    r = (int)ei[wgl];                 // row = ei[0][e]
    c = (int)ei[(size_t)E + wgl];     // col = ei[1][e]
  } else {
    r = c = wgl - E;                  // self loop
  }
  const float norm = dinv[r] * dinv[c];
  const float4 h = *(const float4*)(H + (size_t)r * FEAT + lane * 4);
  float* dst = acc + (size_t)c * FEAT + lane * 4;
  atomicAdd(dst + 0, h.x * norm);
  atomicAdd(dst + 1, h.y * norm);
  atomicAdd(dst + 2, h.z * norm);
  atomicAdd(dst + 3, h.w * norm);
}

// ---------------------------------------------------------------------------
// out = relu(acc + bias[f])
// ---------------------------------------------------------------------------
__global__ void gcn_bias_relu(const float* __restrict__ acc,
                              const float* __restrict__ b,
                              float* __restrict__ out, int total) {
  int i = blockIdx.x * blockDim.x + threadIdx.x;
  if (i < total) {
    float v = acc[i] + b[i & (FEAT - 1)];
    out[i] = v > 0.0f ? v : 0.0f;
  }
}

// ---------------------------------------------------------------------------
// Classifier: out[n][o] = sum_k x[n][k] * Wc[k][o] + bc[o]  (128 -> 10)
// 16 thread slots per node (10 active). Tiny GEMM; scalar is fine.
// ---------------------------------------------------------------------------
__global__ void gcn_classifier(const float* __restrict__ x,
                               const float* __restrict__ Wc,
                               const float* __restrict__ bc,
                               float* __restrict__ out, int n) {
  int t = blockIdx.x * blockDim.x + threadIdx.x;
  int node = t >> 4;
  int o = t & 15;
  if (node < n && o < OUTD) {
    float s = bc[o];
    const float* __restrict__ xr = x + (size_t)node * FEAT;
#pragma unroll 4
    for (int k = 0; k < FEAT; ++k) s += xr[k] * Wc[k * OUTD + o];
    out[(size_t)node * OUTD + o] = s;
  }
}

// ---------------------------------------------------------------------------
// Host-side orchestration
// ---------------------------------------------------------------------------
extern "C" void kernel_launch(void* const* d_in, const int* in_sizes, int n_in,
                              void* d_out, int out_size, void* d_ws, size_t ws_size,
                              hipStream_t stream) {
  const float*     fts  = (const float*)d_in[0];
  const long long* ei   = (const long long*)d_in[1];  // int64 (2, E), row-major
  // d_in[2..5] = W_mlp0/b_mlp0/W_mlp1/b_mlp1: dead branch in reference, skipped.
  const float*     W_g1 = (const float*)d_in[6];
  const float*     b_g1 = (const float*)d_in[7];
  const float*     W_g2 = (const float*)d_in[8];
  const float*     b_g2 = (const float*)d_in[9];
  const float*     W_c  = (const float*)d_in[10];
  const float*     b_c  = (const float*)d_in[11];

  const int N = in_sizes[0] / FEAT;    // 16384 nodes
  const int E = in_sizes[1] / 2;       // 524288 edges
  const int NF = N * FEAT;             // 2,097,152 floats

  // Workspace partition (floats): deg | dinv | H | acc | x1
  float* ws   = (float*)d_ws;
  float* deg  = ws;
  float* dinv = deg + N;
  float* H    = dinv + N;
  float* acc  = H + NF;
  float* x1   = acc + NF;

  // Output layout: (out 16384x10) then (x 16384x128), concatenated flat.
  float* out_logits = (float*)d_out;
  float* out_x      = out_logits + (size_t)N * OUTD;

  const int T = 256;
  const int gN   = (N + T - 1) / T;
  const int gE   = (E + T - 1) / T;
  const int gNF  = (NF + T - 1) / T;
  const int gTile = N / 16;                           // 1024 GEMM blocks
  const int nEdgeWaves = E + N;                       // edges + self loops
  const int gAgg = (nEdgeWaves + 7) / 8;              // 8 waves per block

  // --- normalization ---
  gcn_deg_init <<<gN, T, 0, stream>>>(deg, N);
  gcn_deg_count<<<gE, T, 0, stream>>>(ei, deg, E);
  gcn_dinv     <<<gN, T, 0, stream>>>(deg, dinv, N);

  // --- layer 1: H = fts @ W_g1 ; acc = scatter ; x1 = relu(acc + b_g1) ---
  gcn_gemm128_wmma<<<gTile, T, 0, stream>>>(fts, W_g1, H);
  gcn_zero        <<<gNF, T, 0, stream>>>(acc, NF);
  gcn_aggregate   <<<gAgg, T, 0, stream>>>(H, dinv, ei, acc, E, N);
  gcn_bias_relu   <<<gNF, T, 0, stream>>>(acc, b_g1, x1, NF);

  // --- layer 2: H = x1 @ W_g2 ; acc = scatter ; x = relu(acc + b_g2) ---
  gcn_gemm128_wmma<<<gTile, T, 0, stream>>>(x1, W_g2, H);
  gcn_zero        <<<gNF, T, 0, stream>>>(acc, NF);
  gcn_aggregate   <<<gAgg, T, 0, stream>>>(H, dinv, ei, acc, E, N);
  gcn_bias_relu   <<<gNF, T, 0, stream>>>(acc, b_g2, out_x, NF);

  // --- classifier: logits = x @ W_c + b_c ---
  gcn_classifier<<<(N * 16 + T - 1) / T, T, 0, stream>>>(out_x, W_c, b_c,
                                                         out_logits, N);
}